// SimpleMaskLayer_51960514347299
// MI455X (gfx1250) — compile-verified
//
#include <hip/hip_runtime.h>
#include <hip/hip_bf16.h>

typedef __attribute__((ext_vector_type(2))) float v2f;
typedef __attribute__((ext_vector_type(8))) float v8f;
typedef __attribute__((vector_size(16)))    int   v4i;

#define B_   32
#define S_   4096
#define H_   1024
#define ROWS_ (B_ * S_)          // 131072 GEMV rows
#define TILES_ (ROWS_ / 16)      // 8192 16-row WMMA tiles
#define WAVES_PER_BLOCK 4
#define THREADS_ (WAVES_PER_BLOCK * 32)

#define BK      64                 // K elements staged per chunk
#define NCH     (H_ / BK)          // 16 chunks
#define ASTRIDE 68                 // floats per staged row (64 + 4 pad -> conflict-free b128)
#define TILEF   (16 * ASTRIDE)     // floats per tile buffer (1088)

// ---- CDNA5 async-copy + split-counter waits (builtin if present, else ISA asm) ----
__device__ __forceinline__ void async_copy_b128(const float* g, float* l) {
#if __has_builtin(__builtin_amdgcn_global_load_async_to_lds_b128)
  __builtin_amdgcn_global_load_async_to_lds_b128(
      (__attribute__((address_space(1))) v4i*)g,
      (__attribute__((address_space(3))) v4i*)l, 0, 0);
#else
  unsigned ldsoff = (unsigned)(size_t)(__attribute__((address_space(3))) void*)l;
  asm volatile("global_load_async_to_lds_b128 %0, %1, off"
               :: "v"(ldsoff), "v"(g) : "memory");
#endif
}

#if __has_builtin(__builtin_amdgcn_s_wait_asynccnt)
#define WAIT_ASYNC(n) __builtin_amdgcn_s_wait_asynccnt(n)
#else
#define WAIT_ASYNC(n) asm volatile("s_wait_asynccnt %0" :: "n"(n) : "memory")
#endif

#if __has_builtin(__builtin_amdgcn_s_wait_dscnt)
#define WAIT_DS(n) __builtin_amdgcn_s_wait_dscnt(n)
#else
#define WAIT_DS(n) asm volatile("s_wait_dscnt %0" :: "n"(n) : "memory")
#endif

// ---------------------------------------------------------------------------
// Kernel 1: h[row] = dot(x[row,:], W) + bias.
// Per wave32: one 16-row tile. x is streamed to LDS with coalesced
// GLOBAL_LOAD_ASYNC_TO_LDS_B128 (double-buffered, ASYNCcnt-tracked), then the
// WMMA A operand is gathered from LDS with conflict-free ds_load_b128.
// K axis is permuted (lanes 0-15: kb,kb+1; lanes 16-31: kb+4,kb+5) so one
// b128 read feeds two V_WMMA_F32_16X16X4_F32; W (B operand, broadcast over N)
// uses the identical permutation.
// ---------------------------------------------------------------------------
__global__ __launch_bounds__(THREADS_) void gemv_wmma_kernel(
    const float* __restrict__ x, const float* __restrict__ W,
    const float* __restrict__ bias, float* __restrict__ h) {
  __shared__ float Wl[H_];
  __shared__ float At[WAVES_PER_BLOCK * 2 * TILEF];

  // Stage W (4 KB) into LDS, coalesced b128.
  {
    const float4* W4  = (const float4*)W;
    float4*       Wl4 = (float4*)Wl;
#pragma unroll
    for (int i = threadIdx.x; i < H_ / 4; i += THREADS_) Wl4[i] = W4[i];
  }

  const int wave = threadIdx.x >> 5;
  const int lane = threadIdx.x & 31;
  const int tile = blockIdx.x * WAVES_PER_BLOCK + wave;
  const int row0 = tile << 4;
  const int hi   = lane >> 4;   // half-wave select
  const int lo   = lane & 15;

  // Async staging: per instruction, lanes 0-15 copy 16B segs of row (2i),
  // lanes 16-31 of row (2i+1) -> 256B contiguous per row, fully coalesced.
  const float* gsrc  = x + (size_t)(row0 + hi) * H_ + lo * 4;
  float*       lbase = At + (wave * 2) * TILEF;

  auto issue_chunk = [&](int c) {
    const float* g = gsrc + c * BK;
    float*       l = lbase + (c & 1) * TILEF + hi * ASTRIDE + lo * 4;
#pragma unroll
    for (int i = 0; i < 8; ++i)
      async_copy_b128(g + (size_t)(2 * i) * H_, l + 2 * i * ASTRIDE);
  };

  issue_chunk(0);      // prefetch chunk 0 (overlaps with W-staging barrier)
  __syncthreads();     // Wl visible to all waves

  // Compute-side lane addressing (A row = lo; physical-K offset = 4*hi).
  const float* ar = lbase + lo * ASTRIDE + hi * 4;
  const float* wr = Wl + hi * 4;

  v8f acc = {};
  for (int c = 0; c < NCH; ++c) {
    if (c + 1 < NCH) {
      WAIT_DS(0);            // prior reads of the buffer being overwritten done
      issue_chunk(c + 1);    // prefetch next chunk
      WAIT_ASYNC(8);         // in-order completion => chunk c has landed
    } else {
      WAIT_ASYNC(0);
    }
    const float* a0 = ar + (c & 1) * TILEF;
    const float* w0 = wr + c * BK;
#pragma unroll
    for (int kb = 0; kb < BK; kb += 8) {
      float4 a4 = *(const float4*)(a0 + kb);   // ds_load_b128 (2-pass, optimal)
      float4 w4 = *(const float4*)(w0 + kb);   // ds_load_b128 (broadcast)
      v2f aA = {a4.x, a4.y}, aB = {a4.z, a4.w};
      v2f wA = {w4.x, w4.y}, wB = {w4.z, w4.w};
      acc = __builtin_amdgcn_wmma_f32_16x16x4_f32(
          false, aA, false, wA, (short)0, acc, false, false);
      acc = __builtin_amdgcn_wmma_f32_16x16x4_f32(
          false, aB, false, wB, (short)0, acc, false, false);
    }
  }

  const float bv    = bias[0];
  const int   mbase = hi << 3;  // D VGPR r holds M=r (lanes 0-15) / M=8+r
#pragma unroll
  for (int r = 0; r < 8; ++r) {
    if (lo == r) h[row0 + mbase + r] = acc[r] + bv;  // lanes r, r+16 -> 16 rows
  }
}

// ---------------------------------------------------------------------------
// Kernel 2: per-batch masked min/max of h (mean/std cancel algebraically).
// ---------------------------------------------------------------------------
__global__ __launch_bounds__(256) void rowminmax_kernel(
    const float* __restrict__ h, const unsigned char* __restrict__ mask,
    float* __restrict__ mn, float* __restrict__ mx) {
  __shared__ float smn[256];
  __shared__ float smx[256];
  const int b = blockIdx.x;
  const int t = threadIdx.x;
  const float*         hr = h + (size_t)b * S_;
  const unsigned char* mr = mask + (size_t)b * S_;

  float lmn = INFINITY, lmx = -INFINITY;
#pragma unroll 4
  for (int s = t; s < S_; s += 256) {
    float v = hr[s];
    if (mr[s]) { lmn = fminf(lmn, v); lmx = fmaxf(lmx, v); }
  }
  smn[t] = lmn; smx[t] = lmx;
  __syncthreads();
#pragma unroll
  for (int off = 128; off > 0; off >>= 1) {
    if (t < off) {
      smn[t] = fminf(smn[t], smn[t + off]);
      smx[t] = fmaxf(smx[t], smx[t + off]);
    }
    __syncthreads();
  }
  if (t == 0) { mn[b] = smn[0]; mx[b] = smx[0]; }
}

// ---------------------------------------------------------------------------
// Kernel 3: out = mask ? (h - hmin) / (hmax - hmin) : 0
// ---------------------------------------------------------------------------
__global__ __launch_bounds__(256) void finalize_kernel(
    const float* __restrict__ h, const unsigned char* __restrict__ mask,
    const float* __restrict__ mn, const float* __restrict__ mx,
    float* __restrict__ out) {
  const int i = blockIdx.x * 256 + threadIdx.x;
  const int b = i >> 12;  // / S_
  const float lo = mn[b];
  const float r  = 1.0f / (mx[b] - lo);
  out[i] = mask[i] ? (h[i] - lo) * r : 0.0f;
}

extern "C" void kernel_launch(void* const* d_in, const int* in_sizes, int n_in,
                              void* d_out, int out_size, void* d_ws, size_t ws_size,
                              hipStream_t stream) {
  const float*         x    = (const float*)d_in[0];          // [B,S,H] fp32
  const unsigned char* mask = (const unsigned char*)d_in[1];  // [B,S] bool (1B)
  const float*         W    = (const float*)d_in[2];          // [H] fp32
  const float*         bias = (const float*)d_in[3];          // [1] fp32
  float*               out  = (float*)d_out;                  // [B,S,1] fp32

  // Workspace layout: h[ROWS_] | mn[B_] | mx[B_]
  float* h  = (float*)d_ws;
  float* mn = h + ROWS_;
  float* mx = mn + B_;

  gemv_wmma_kernel<<<TILES_ / WAVES_PER_BLOCK, THREADS_, 0, stream>>>(x, W, bias, h);
  rowminmax_kernel<<<B_, 256, 0, stream>>>(h, mask, mn, mx);
  finalize_kernel<<<ROWS_ / 256, 256, 0, stream>>>(h, mask, mn, mx, out);
}